// KnowledgeEmbedding_38019050504968
// MI455X (gfx1250) — compile-verified
//
#include <hip/hip_runtime.h>
#include <math.h>
#include <stdint.h>

// ---------------------------------------------------------------------------
// KnowledgeEmbedding loss on gfx1250 (MI455X).
//   B=16384 rows, D=128, S=100 negatives, 8 relations.
//   neg_logits GEMM (B x 128) @ (128 x S): V_WMMA_F32_16X16X4_F32.
//   nv staged in LDS via GLOBAL_LOAD_ASYNC_TO_LDS_B128 (ASYNCcnt path).
//   Positive loss + ||hv||^2/||tv||^2 fused into the A-fragment load loop.
// ---------------------------------------------------------------------------

typedef __attribute__((ext_vector_type(2))) float v2f;
typedef __attribute__((ext_vector_type(8))) float v8f;

#define B_ROWS     16384
#define DIM        128
#define NS         100
#define NTILES     7            // ceil(100/16)
#define MTILES     1024         // B_ROWS / 16
#define LDS_STRIDE 132          // floats; 132 % 64 == 4 -> conflict-free b64 reads
#define L2REG      1e-5f

// stable softplus(x) = log(1+exp(x)) = max(x,0) + log1p(exp(-|x|))
__device__ __forceinline__ float softplus_f(float x) {
    return fmaxf(x, 0.0f) + log1pf(expf(-fabsf(x)));
}

__device__ __forceinline__ float waveReduceSum(float v) {
#pragma unroll
    for (int m = 16; m >= 1; m >>= 1) v += __shfl_xor(v, m, 32);
    return v;
}

struct RelParams {
    const float* head[8];
    const float* tail[8];
    const float* bias[8];
    const int*   neg[8];
    int          hcol[8];
    int          tcol[8];
    const float* rvecs;   // relation r uses rvecs + r*DIM
    const int*   bidx;    // (B,8)
};

// ---------------------------------------------------------------------------
// Main kernel: grid = (MTILES/8, 8 relations), 256 threads (8 waves).
// Wave w of block handles M-tile = blockIdx.x*8 + w (16 rows).
// Produces per-tile partials: loss (pos+neg), ||hv||^2, ||tv||^2.
// ---------------------------------------------------------------------------
__global__ void __launch_bounds__(256)
relKernel(RelParams p,
          float* __restrict__ lossPart,   // 8 * MTILES
          float* __restrict__ hssPart,    // 8 * MTILES
          float* __restrict__ tssPart)    // 8 * MTILES
{
    const int rel = blockIdx.y;
    const float* __restrict__ headTab = p.head[rel];
    const float* __restrict__ tailTab = p.tail[rel];
    const float* __restrict__ biasTab = p.bias[rel];
    const int*   __restrict__ negIdx  = p.neg[rel];
    const float* __restrict__ rvec    = p.rvecs + rel * DIM;
    const int*   __restrict__ bidx    = p.bidx;
    const int hcol = p.hcol[rel], tcol = p.tcol[rel];

    __shared__ float snv[NS * LDS_STRIDE];   // 100 x 128 fp32, padded rows

    // ---- stage nv into LDS with CDNA5 async loads (ASYNCcnt) ----
    // 100 rows * 32 float4 = 3200 b128 transfers across 256 threads.
    for (int e = threadIdx.x; e < NS * (DIM / 4); e += 256) {
        const int row = e >> 5;                 // float4 index / 32
        const int c4  = (e & 31) * 4;           // float column (16B aligned)
        const float* gp = tailTab + (size_t)negIdx[row] * DIM + c4;
        const unsigned ldsOff =
            (unsigned)(uintptr_t)(&snv[row * LDS_STRIDE + c4]);
        asm volatile("global_load_async_to_lds_b128 %0, %1, off"
                     :: "v"(ldsOff), "v"(gp) : "memory");
    }

    const int lane = threadIdx.x & 31;
    const int wid  = threadIdx.x >> 5;
    const int tile = blockIdx.x * 8 + wid;     // 0..1023
    const int Mbase = tile * 16;
    const int half  = lane >> 4;
    const int lp    = lane & 15;

    // ---- A fragments (ev = hv + rvec) + fused pos/norm partials ----
    // Lane (lp,half) owns cols {4k+2*half, 4k+2*half+1} of row Mbase+lp.
    const int h = bidx[(Mbase + lp) * 8 + hcol];
    const int t = bidx[(Mbase + lp) * 8 + tcol];
    const float* __restrict__ hrow = headTab + (size_t)h * DIM;
    const float* __restrict__ trow = tailTab + (size_t)t * DIM;

    v2f a[32];
    float dotp = 0.f, hssp = 0.f, tssp = 0.f;
#pragma unroll
    for (int k = 0; k < 32; ++k) {
        const int col = 4 * k + 2 * half;      // even -> 8B aligned
        const float h0 = hrow[col], h1 = hrow[col + 1];
        const float r0 = rvec[col], r1 = rvec[col + 1];
        const float t0 = trow[col], t1 = trow[col + 1];
        v2f f; f.x = h0 + r0; f.y = h1 + r1;
        a[k] = f;
        dotp += f.x * t0 + f.y * t1;           // partial ev.tv (half the cols)
        hssp += h0 * h0 + h1 * h1;
        tssp += t0 * t0 + t1 * t1;
    }

    // positive loss for row Mbase+lp: combine the two column halves.
    const float fulldot = dotp + __shfl_xor(dotp, 16, 32);
    const float bl = biasTab[t];
    float local = (half == 0) ? softplus_f(-(fulldot + bl)) : 0.0f;

    // positive-tail bias for the 8 C rows this lane owns (C: M = r + 8*half)
    float bvals[8];
#pragma unroll
    for (int r = 0; r < 8; ++r)
        bvals[r] = biasTab[bidx[(Mbase + r + 8 * half) * 8 + tcol]];

    // ---- wait for async staging, then sync the workgroup ----
    asm volatile("s_wait_asynccnt 0x0" ::: "memory");
    __syncthreads();

    // ---- negative logits: 7 N-tiles x 32 fp32 WMMA (K=4 each) ----
#pragma unroll 1
    for (int nt = 0; nt < NTILES; ++nt) {
        const int jj  = nt * 16 + lp;
        const int jc  = (jj < NS) ? jj : (NS - 1);     // clamp (select)
        const float msk = (jj < NS) ? 1.0f : 0.0f;     // mask (select)
        const float* __restrict__ srow = &snv[jc * LDS_STRIDE];

        v8f acc = {0.f, 0.f, 0.f, 0.f, 0.f, 0.f, 0.f, 0.f};
#pragma unroll
        for (int k = 0; k < 32; ++k) {
            const int col = 4 * k + 2 * half;          // B frag mirrors A
            v2f b;
            b.x = srow[col];
            b.y = srow[col + 1];
            acc = __builtin_amdgcn_wmma_f32_16x16x4_f32(
                      false, a[k], false, b, (short)0, acc, false, false);
        }
#pragma unroll
        for (int r = 0; r < 8; ++r)
            local += msk * softplus_f(acc[r] + bvals[r]);
    }

    const float ls = waveReduceSum(local);
    const float hs = waveReduceSum(hssp);
    const float ts = waveReduceSum(tssp);
    if (lane == 0) {
        lossPart[rel * MTILES + tile] = ls;
        hssPart [rel * MTILES + tile] = hs;
        tssPart [rel * MTILES + tile] = ts;
    }
}

// ---------------------------------------------------------------------------
// ||nv||^2, one block per relation.
// ---------------------------------------------------------------------------
__global__ void __launch_bounds__(256)
nvssKernel(RelParams p, float* __restrict__ out /* 8 */)
{
    const int rel = blockIdx.x;
    const float* __restrict__ tailTab = p.tail[rel];
    const int*   __restrict__ negIdx  = p.neg[rel];

    float ss = 0.f;
    for (int e = threadIdx.x; e < NS * DIM; e += 256) {
        const int jrow = e >> 7;
        const int col  = e & (DIM - 1);
        const float v = tailTab[(size_t)negIdx[jrow] * DIM + col];
        ss += v * v;
    }
    __shared__ float sm[256];
    const int tid = threadIdx.x;
    sm[tid] = ss;
    __syncthreads();
#pragma unroll
    for (int s = 128; s > 0; s >>= 1) {
        if (tid < s) sm[tid] += sm[tid + s];
        __syncthreads();
    }
    if (tid == 0) out[rel] = sm[0];
}

// ---------------------------------------------------------------------------
// Deterministic finalize: fixed-order strided sums + LDS tree.
// ---------------------------------------------------------------------------
__global__ void __launch_bounds__(256)
finalizeKernel(const float* __restrict__ lossPart,  // 8*1024
               const float* __restrict__ hssPart,   // 8*1024
               const float* __restrict__ tssPart,   // 8*1024
               const float* __restrict__ nvss,      // 8
               float*       __restrict__ out)
{
    __shared__ float sm[256];
    const int tid = threadIdx.x;

    float v = 0.f;
    for (int i = tid; i < 8 * MTILES; i += 256) v += lossPart[i];
    sm[tid] = v;
    __syncthreads();
#pragma unroll
    for (int s = 128; s > 0; s >>= 1) {
        if (tid < s) sm[tid] += sm[tid + s];
        __syncthreads();
    }
    const float lossSum = sm[0];
    __syncthreads();

    float reg = 0.f;
    for (int rel = 0; rel < 8; ++rel) {
        float a = 0.f, b = 0.f;
        for (int i = tid; i < MTILES; i += 256) {
            a += hssPart[rel * MTILES + i];
            b += tssPart[rel * MTILES + i];
        }
        sm[tid] = a;
        __syncthreads();
#pragma unroll
        for (int s = 128; s > 0; s >>= 1) {
            if (tid < s) sm[tid] += sm[tid + s];
            __syncthreads();
        }
        const float hs = sm[0];
        __syncthreads();

        sm[tid] = b;
        __syncthreads();
#pragma unroll
        for (int s = 128; s > 0; s >>= 1) {
            if (tid < s) sm[tid] += sm[tid + s];
            __syncthreads();
        }
        const float ts = sm[0];
        __syncthreads();

        reg += sqrtf(hs) + sqrtf(ts) + sqrtf(nvss[rel]);
    }

    if (tid == 0)
        out[0] = lossSum / (float)B_ROWS + L2REG * reg;
}

// ---------------------------------------------------------------------------
extern "C" void kernel_launch(void* const* d_in, const int* in_sizes, int n_in,
                              void* d_out, int out_size, void* d_ws, size_t ws_size,
                              hipStream_t stream)
{
    (void)in_sizes; (void)n_in; (void)out_size; (void)ws_size;

    const float* user  = (const float*)d_in[0];
    const float* prod  = (const float*)d_in[1];
    const float* word  = (const float*)d_in[2];
    const float* relp  = (const float*)d_in[3];
    const float* brand = (const float*)d_in[4];
    const float* cat   = (const float*)d_in[5];

    RelParams p;
    const float* heads[8] = { user, user, prod, prod, prod, prod, prod, prod };
    const float* tails[8] = { prod, word, word, brand, cat, relp, relp, relp };
    const int    hcols[8] = { 0, 0, 1, 1, 1, 1, 1, 1 };
    const int    tcols[8] = { 1, 2, 2, 3, 4, 5, 6, 7 };
    for (int r = 0; r < 8; ++r) {
        p.head[r] = heads[r];
        p.tail[r] = tails[r];
        p.bias[r] = (const float*)d_in[7 + r];
        p.neg[r]  = (const int*)  d_in[16 + r];
        p.hcol[r] = hcols[r];
        p.tcol[r] = tcols[r];
    }
    p.rvecs = (const float*)d_in[6];
    p.bidx  = (const int*)  d_in[15];

    // workspace layout (floats)
    float* ws       = (float*)d_ws;
    float* lossPart = ws;                     // 8 * 1024
    float* hssPart  = lossPart + 8 * MTILES;  // 8 * 1024
    float* tssPart  = hssPart  + 8 * MTILES;  // 8 * 1024
    float* nvssArr  = tssPart  + 8 * MTILES;  // 8

    relKernel<<<dim3(MTILES / 8, 8), 256, 0, stream>>>(p, lossPart, hssPart, tssPart);
    nvssKernel<<<8, 256, 0, stream>>>(p, nvssArr);
    finalizeKernel<<<1, 256, 0, stream>>>(lossPart, hssPart, tssPart, nvssArr,
                                          (float*)d_out);
}